// VarMaskedRNN_78194174591226
// MI455X (gfx1250) — compile-verified
//
#include <hip/hip_runtime.h>
#include <hip/hip_bf16.h>
#include <math.h>

// ---------------------------------------------------------------------------
// VarMaskedRNN for MI455X (gfx1250): fp32 WMMA (V_WMMA_F32_16X16X4_F32)
//   B=32, T=1024, IN=1024, H=512, L=2, D=2
// Phase 1: xin[c] = (src * noise_in[c]) @ w_ih[c]^T + b_ih[c]
//          WMMA GEMM, B-panel staged in LDS via async global->LDS copies
//          (double buffered, ASYNCcnt), conflict-free padded LDS layout.
// Phase 2: sequential masked scan, one workgroup per direction, h in LDS
//          with XOR bank swizzle (conflict-free A-fragment reads), WMMA.
// ---------------------------------------------------------------------------

typedef __attribute__((ext_vector_type(2))) float v2f;
typedef __attribute__((ext_vector_type(8))) float v8f;

#define RNN_B   32
#define RNN_T   1024
#define RNN_IN  1024
#define RNN_H   512
#define KC      64          // GEMM K-chunk staged in LDS
#define BPAD    68          // padded row (dwords): (4*lane + k) % 64 distinct

static __device__ __forceinline__ v8f wmma_f32_k4(v2f a, v2f b, v8f c) {
  // D = A(16x4,f32) * B(4x16,f32) + C(16x16,f32)
  return __builtin_amdgcn_wmma_f32_16x16x4_f32(
      /*neg_a=*/false, a, /*neg_b=*/false, b,
      /*c_mod=*/(short)0, c, /*reuse_a=*/false, /*reuse_b=*/false);
}

// Low 32 bits of a flat shared-aperture address == LDS byte offset.
static __device__ __forceinline__ unsigned lds_off_of(const void* p) {
  return (unsigned)(unsigned long long)p;
}

// ---------------------------------------------------------------------------
// Input-projection GEMM: dst[r=(t,b)][n] = sum_k src(t,b,k)*noise(b,k)*w(n,k)+bias(n)
//   M = T*B = 32768, N = 512, K = 1024.
//   256 threads = 8 waves; wave computes 16(M) x 64(N); grid = (256, 8).
//   w panel (64 N-rows x 64 K) staged in LDS with async copies, reused by all
//   8 waves (cuts global W traffic 8x in a compute-bound kernel).
// ---------------------------------------------------------------------------
__global__ __launch_bounds__(256) void xin_gemm_kernel(
    const float* __restrict__ src, long strideT, long strideB,
    const float* __restrict__ noise,   // [B, K]
    const float* __restrict__ w,       // [N=512, K=1024] row-major
    const float* __restrict__ bias,    // [N]
    float* __restrict__ dst)           // [T*B, 512]
{
  const int tid    = threadIdx.x;
  const int lane   = tid & 31;
  const int wave   = tid >> 5;
  const int lane15 = lane & 15;
  const int half   = lane >> 4;          // 0 or 1

  const int mrow = blockIdx.x * 128 + wave * 16;
  const int ncol = blockIdx.y * 64;

  __shared__ float ldsB[2][64 * BPAD];   // 2 x 17.4 KB double buffer

  // A-fragment row for this lane
  const int mr = mrow + lane15;
  const int t  = mr >> 5;                // row = t*B + b, B = 32
  const int b  = mr & 31;
  const float* arow = src + (long)t * strideT + (long)b * strideB;
  const float* nrow = noise + (long)b * RNN_IN;

  // cooperative staging: thread -> (row, 16-byte column) of the 64x64 panel
  const int srow0 = tid >> 4;            // 0..15 (then +16*i)
  const int scol  = (tid & 15) * 4;      // float offset within K-chunk
  const float* wpanel = w + (long)(ncol + srow0) * RNN_IN + scol;
  const unsigned ldsBase[2] = { lds_off_of(&ldsB[0][0]), lds_off_of(&ldsB[1][0]) };

  auto stage = [&](int kc, int pp) {
    const unsigned lb = ldsBase[pp];
    #pragma unroll
    for (int i = 0; i < 4; ++i) {
      const float* g = wpanel + (long)i * 16 * RNN_IN + kc;
      const unsigned l = lb + (unsigned)(((srow0 + i * 16) * BPAD + scol) * 4);
      asm volatile("global_load_async_to_lds_b128 %0, %1, off"
                   :: "v"(l), "v"(g) : "memory");
    }
  };

  v8f acc[4];
  #pragma unroll
  for (int i = 0; i < 4; ++i) acc[i] = (v8f)0.0f;

  stage(0, 0);
  asm volatile("s_wait_asynccnt 0x0" ::: "memory");
  __syncthreads();

  int p = 0;
  for (int kc = 0; kc < RNN_IN; kc += KC) {
    if (kc + KC < RNN_IN) stage(kc + KC, p ^ 1);   // prefetch next panel

    const float* bbase = &ldsB[p][lane15 * BPAD];
    #pragma unroll 4
    for (int k2 = 0; k2 < KC; k2 += 4) {
      const int k0g = kc + k2 + 2 * half;          // global k for A frag
      const int k0l = k2 + 2 * half;               // LDS col for B frag
      v2f av = *(const v2f*)(arow + k0g);
      v2f nv = *(const v2f*)(nrow + k0g);
      av *= nv;                                    // variational input dropout
      v2f b0 = *(const v2f*)(bbase + k0l);
      v2f b1 = *(const v2f*)(bbase + 16 * BPAD + k0l);
      v2f b2 = *(const v2f*)(bbase + 32 * BPAD + k0l);
      v2f b3 = *(const v2f*)(bbase + 48 * BPAD + k0l);
      acc[0] = wmma_f32_k4(av, b0, acc[0]);
      acc[1] = wmma_f32_k4(av, b1, acc[1]);
      acc[2] = wmma_f32_k4(av, b2, acc[2]);
      acc[3] = wmma_f32_k4(av, b3, acc[3]);
    }

    asm volatile("s_wait_asynccnt 0x0" ::: "memory");  // next panel landed
    __syncthreads();
    p ^= 1;
  }

  // C/D layout: VGPR v -> M = v + 8*half, N = lane15 (within 16x16 tile)
  #pragma unroll
  for (int nt = 0; nt < 4; ++nt) {
    const int n  = ncol + nt * 16 + lane15;
    const float bv = bias[n];
    #pragma unroll
    for (int v = 0; v < 8; ++v) {
      const int rr = mrow + v + 8 * half;
      dst[(long)rr * RNN_H + n] = acc[nt][v] + bv;
    }
  }
}

// ---------------------------------------------------------------------------
// Recurrent scan: one workgroup per direction (grid.x = 2), 1024 thr = 32 waves.
//   h[32][512] in LDS with XOR swizzle: element (b,n) stored at column
//   n ^ ((b&15)<<2).  Swizzle touches bits 2..5 only, so 8-byte v2f accesses
//   stay aligned and A-fragment ds_load_b64 banks = (k0 ^ 4*lane15) -> 0 conflicts.
//   wave w owns N-tile w and both M-tiles; each whh B-fragment feeds 2 WMMAs.
// ---------------------------------------------------------------------------
__global__ __launch_bounds__(1024) void rnn_scan_kernel(
    const float* __restrict__ xin0,    // [T, B, H] cell (dir 0)
    const float* __restrict__ xin1,    // [T, B, H] cell (dir 1)
    const float* __restrict__ whh,     // [4, H, H]
    const float* __restrict__ bhh,     // [4, H]
    const float* __restrict__ noise_h, // [4, B, H]
    const float* __restrict__ mask,    // [B, T]
    float* __restrict__ dst,           // hs out: t*dstT + b*dstB + dir*H + n
    long dstT, long dstB,
    float* __restrict__ hn_base,       // d_out + B*T*2H ; + c*B*H per cell
    int cellBase)
{
  const int dir  = blockIdx.x;               // 0 = forward, 1 = backward
  const int c    = cellBase + dir;
  const int tid  = threadIdx.x;
  const int lane = tid & 31;
  const int wave = tid >> 5;                 // 0..31 == N-tile index
  const int lane15 = lane & 15;
  const int half   = lane >> 4;

  const float* xin = (dir == 0) ? xin0 : xin1;
  const float* W   = whh     + (long)c * RNN_H * RNN_H;
  const float* nh  = noise_h + (long)c * RNN_B * RNN_H;
  const float* bh  = bhh     + (long)c * RNN_H;
  float*       hno = hn_base + (long)c * RNN_B * RNN_H;

  __shared__ float h_s[RNN_B][RNN_H];        // 64 KB hidden state (swizzled cols)

  for (int i = tid; i < RNN_B * RNN_H; i += 1024)
    ((float*)h_s)[i] = 0.0f;
  __syncthreads();

  const int n = wave * 16 + lane15;          // output column of this lane
  const float* wrow = W + (long)n * RNN_H;   // B-frag: (k,n) = W[n*512+k]
  const float* hrow0 = &h_s[lane15][0];      // M-tile 0 row (b = lane15)
  const float* hrow1 = &h_s[16 + lane15][0]; // M-tile 1 row (b = 16+lane15)
  const unsigned sw  = (unsigned)lane15 << 2;           // swizzle (same both tiles)
  const float* nh0  = nh + (long)lane15 * RNN_H;
  const float* nh1  = nh + (long)(16 + lane15) * RNN_H;
  const float  bv   = bh[n];

  for (int s = 0; s < RNN_T; ++s) {
    const int t = dir ? (RNN_T - 1 - s) : s;

    v8f acc[2];
    acc[0] = (v8f)0.0f;
    acc[1] = (v8f)0.0f;

    #pragma unroll 4
    for (int k = 0; k < RNN_H; k += 4) {
      const int k0 = k + 2 * half;
      const int kS = (int)((unsigned)k0 ^ sw);           // swizzled LDS column
      v2f bf = *(const v2f*)(wrow + k0);
      v2f a0 = *(const v2f*)(hrow0 + kS) * *(const v2f*)(nh0 + k0);
      v2f a1 = *(const v2f*)(hrow1 + kS) * *(const v2f*)(nh1 + k0);
      acc[0] = wmma_f32_k4(a0, bf, acc[0]);
      acc[1] = wmma_f32_k4(a1, bf, acc[1]);
    }

    const float* xt = xin + (long)t * (RNN_B * RNN_H);
    float hout[16];
    #pragma unroll
    for (int mt = 0; mt < 2; ++mt) {
      #pragma unroll
      for (int v = 0; v < 8; ++v) {
        const int b  = mt * 16 + v + 8 * half;           // C/D layout row
        const int nS = (int)((unsigned)n ^ ((unsigned)(b & 15) << 2));
        const float g    = acc[mt][v] + xt[(long)b * RNN_H + n] + bv;
        const float hnew = tanhf(g);
        const float mval = mask[(long)b * RNN_T + t];
        const float hold = h_s[b][nS];
        hout[mt * 8 + v] = hnew * mval + hold * (1.0f - mval);
      }
    }

    __syncthreads();   // everyone done reading old h
    #pragma unroll
    for (int mt = 0; mt < 2; ++mt) {
      #pragma unroll
      for (int v = 0; v < 8; ++v) {
        const int b  = mt * 16 + v + 8 * half;
        const int nS = (int)((unsigned)n ^ ((unsigned)(b & 15) << 2));
        const float hv = hout[mt * 8 + v];
        h_s[b][nS] = hv;
        dst[(long)t * dstT + (long)b * dstB + (long)dir * RNN_H + n] = hv;
      }
    }
    __syncthreads();   // new h visible before next step
  }

  // final carry -> hn[c]
  #pragma unroll
  for (int mt = 0; mt < 2; ++mt) {
    #pragma unroll
    for (int v = 0; v < 8; ++v) {
      const int b  = mt * 16 + v + 8 * half;
      const int nS = (int)((unsigned)n ^ ((unsigned)(b & 15) << 2));
      hno[(long)b * RNN_H + n] = h_s[b][nS];
    }
  }
}

// ---------------------------------------------------------------------------
// Host launcher
// Inputs: 0:x[B,T,IN] 1:mask[B,T] 2:w_ih[4,H,IN] 3:w_hh[4,H,H]
//         4:b_ih[4,H] 5:b_hh[4,H] 6:noise_in[4,B,IN] 7:noise_h[4,B,H]
// Output: out[B,T,2H] ++ hn[4,B,H]
// Workspace: xin0, xin1 = 2 * T*B*H floats (128 MB).
// Layer-0 seq [T,B,2H] is staged inside d_out (overwritten by layer 1 later).
// ---------------------------------------------------------------------------
extern "C" void kernel_launch(void* const* d_in, const int* in_sizes, int n_in,
                              void* d_out, int out_size, void* d_ws, size_t ws_size,
                              hipStream_t stream) {
  const float* x        = (const float*)d_in[0];
  const float* mask     = (const float*)d_in[1];
  const float* w_ih     = (const float*)d_in[2];
  const float* w_hh     = (const float*)d_in[3];
  const float* b_ih     = (const float*)d_in[4];
  const float* b_hh     = (const float*)d_in[5];
  const float* noise_in = (const float*)d_in[6];
  const float* noise_h  = (const float*)d_in[7];

  float* out = (float*)d_out;
  float* ws  = (float*)d_ws;

  const long xinElems = (long)RNN_T * RNN_B * RNN_H;   // 16M floats
  float* xin0 = ws;
  float* xin1 = ws + xinElems;
  float* seq  = out;                                    // [T, B, 2H] staging
  float* hn   = out + (long)RNN_B * RNN_T * (2 * RNN_H);

  const long cellW_ih = (long)RNN_H * RNN_IN;
  const long cellNin  = (long)RNN_B * RNN_IN;

  dim3 ggrid(256, 8), gblock(256);

  // ---- Layer 0: input projections from x [B,T,IN] ----
  xin_gemm_kernel<<<ggrid, gblock, 0, stream>>>(
      x, /*strideT=*/(long)RNN_IN, /*strideB=*/(long)RNN_T * RNN_IN,
      noise_in + 0 * cellNin, w_ih + 0 * cellW_ih, b_ih + 0 * RNN_H, xin0);
  xin_gemm_kernel<<<ggrid, gblock, 0, stream>>>(
      x, (long)RNN_IN, (long)RNN_T * RNN_IN,
      noise_in + 1 * cellNin, w_ih + 1 * cellW_ih, b_ih + 1 * RNN_H, xin1);

  // ---- Layer 0 scan -> seq (staged in d_out as [T,B,2H]) ----
  rnn_scan_kernel<<<dim3(2), dim3(1024), 0, stream>>>(
      xin0, xin1, w_hh, b_hh, noise_h, mask,
      seq, /*dstT=*/(long)RNN_B * 2 * RNN_H, /*dstB=*/(long)2 * RNN_H,
      hn, /*cellBase=*/0);

  // ---- Layer 1: input projections from seq [T,B,2H] (2H == IN == 1024) ----
  xin_gemm_kernel<<<ggrid, gblock, 0, stream>>>(
      seq, /*strideT=*/(long)RNN_B * 2 * RNN_H, /*strideB=*/(long)2 * RNN_H,
      noise_in + 2 * cellNin, w_ih + 2 * cellW_ih, b_ih + 2 * RNN_H, xin0);
  xin_gemm_kernel<<<ggrid, gblock, 0, stream>>>(
      seq, (long)RNN_B * 2 * RNN_H, (long)2 * RNN_H,
      noise_in + 3 * cellNin, w_ih + 3 * cellW_ih, b_ih + 3 * RNN_H, xin1);

  // ---- Layer 1 scan -> final output [B,T,2H] ----
  rnn_scan_kernel<<<dim3(2), dim3(1024), 0, stream>>>(
      xin0, xin1, w_hh, b_hh, noise_h, mask,
      out, /*dstT=*/(long)2 * RNN_H, /*dstB=*/(long)RNN_T * 2 * RNN_H,
      hn, /*cellBase=*/2);
}